// PhotometricLoss_13804024889843
// MI455X (gfx1250) — compile-verified
//
#include <hip/hip_runtime.h>

// Photometric loss (warp + SSIM + L1 + masked mean), fused single pass.
// Memory-bound: ~206 MB read -> ~9us floor at 23.3 TB/s. Uses gfx1250 async
// global->LDS DMA (ASYNCcnt path) to overlap left-tile streaming with the
// disparity-warp gathers.

#define BATCH 8
#define CH    3
#define IMH   720
#define IMW   1280
#define TH    16
#define TW    64
#define LW    72           // LDS row stride (floats): keeps interior 16B-aligned
#define HR    (TH + 2)     // 18 halo rows
#define HC    (TW + 2)     // 66 halo cols
#define NHALO (HR * HC)    // 1188
#define NT    256

__global__ __launch_bounds__(64) void photo_init(double* ws) {
  if (threadIdx.x < 2) ws[threadIdx.x] = 0.0;
}

__global__ __launch_bounds__(NT) void photo_main(const float* __restrict__ disp,
                                                 const float* __restrict__ left,
                                                 const float* __restrict__ right,
                                                 double* __restrict__ ws) {
  __shared__ __align__(16) float xs[HR * LW];  // left tile (+halo, zero-padded)
  __shared__ float ysb[HR * LW];               // warped right tile (+halo)
  __shared__ float frs[HR * LW];               // bilinear frac weight
  __shared__ int   i0s[HR * LW];               // gather base col (-1 = out of image)
  __shared__ float vs[TH * TW];                // valid mask (interior)
  __shared__ float pr[NT];
  __shared__ float vr[NT];

  const int tid = threadIdx.x;
  const int w0  = blockIdx.x * TW;
  const int h0  = blockIdx.y * TH;
  const int b   = blockIdx.z;

  float th_photo = 0.f;
  float th_valid = 0.f;

  // ---- Phase 0: warp parameters + valid mask from disp (shared by 3 channels)
  for (int idx = tid; idx < NHALO; idx += NT) {
    const int r = idx / HC, cc = idx % HC;
    const int lp = r * LW + cc + 3;
    const int h = h0 + r - 1, w = w0 + cc - 1;
    float frv = 0.f, validv = 0.f;
    int   i0v = -1;
    if (h >= 0 && h < IMH && w >= 0 && w < IMW) {
      const float d = disp[(size_t)(b * IMH + h) * IMW + w];
      const float xsamp = (float)w - d;
      validv = (xsamp > 0.f && xsamp < (float)(IMW - 1)) ? 1.f : 0.f;
      const float xcl = fminf(fmaxf(xsamp, 0.f), (float)(IMW - 1));
      const float x0  = floorf(xcl);
      frv = xcl - x0;
      i0v = (int)x0;
    }
    frs[lp] = frv;
    i0s[lp] = i0v;
    if (r >= 1 && r <= TH && cc >= 1 && cc <= TW) {
      vs[(r - 1) * TW + (cc - 1)] = validv;
      th_valid += validv;  // each interior pixel counted exactly once
    }
  }

  for (int c = 0; c < CH; ++c) {
    __syncthreads();  // phase-0 LDS visible (c==0); prior stencil reads done (c>0)

    const float* lbase = left  + (size_t)((b * CH + c) * IMH) * IMW;
    const float* rbase = right + (size_t)((b * CH + c) * IMH) * IMW;

    // ---- Async DMA: interior 16x64 left tile -> LDS (one b128 per lane).
    {
      const int r = tid >> 4, k = tid & 15;
      const unsigned voff  = (unsigned)((((h0 + r) * IMW) + w0 + 4 * k) * 4);
      const unsigned laddr = (unsigned)(unsigned long long)(void*)&xs[(r + 1) * LW + 4 + 4 * k];
      asm volatile("global_load_async_to_lds_b128 %0, %1, %2"
                   :: "v"(laddr), "v"(voff), "s"((unsigned long long)lbase)
                   : "memory");
    }

    // ---- Meanwhile: halo frame of left tile + full warped-right tile (gathers).
    for (int idx = tid; idx < NHALO; idx += NT) {
      const int r = idx / HC, cc = idx % HC;
      const int lp = r * LW + cc + 3;
      const int h = h0 + r - 1, w = w0 + cc - 1;
      const bool inb = (h >= 0) && (h < IMH) && (w >= 0) && (w < IMW);
      if (r == 0 || r == HR - 1 || cc == 0 || cc == HC - 1)
        xs[lp] = inb ? lbase[(size_t)h * IMW + w] : 0.f;  // zero-pad for avg_pool
      float y = 0.f;
      const int i0 = i0s[lp];
      if (i0 >= 0) {
        const float f = frs[lp];
        const float* rrow = rbase + (size_t)h * IMW;
        const int i1 = min(i0 + 1, IMW - 1);
        const float g0 = rrow[i0], g1 = rrow[i1];
        y = fmaf(f, g1 - g0, g0);  // (1-f)*g0 + f*g1
      }
      ysb[lp] = y;
    }

    asm volatile("s_wait_asynccnt 0" ::: "memory");
    __syncthreads();

    // ---- 3x3 zero-padded box stats -> SSIM + L1, masked accumulation.
    #pragma unroll
    for (int k = 0; k < 4; ++k) {
      const int p  = k * NT + tid;
      const int ro = p >> 6, co = p & 63;
      float sx = 0.f, sy = 0.f, sxx = 0.f, syy = 0.f, sxy = 0.f, xc = 0.f, yc = 0.f;
      #pragma unroll
      for (int dr = 0; dr < 3; ++dr) {
        #pragma unroll
        for (int dc = 0; dc < 3; ++dc) {
          const int q = (ro + dr) * LW + co + 3 + dc;
          const float xv = xs[q], yv = ysb[q];
          sx += xv; sy += yv;
          sxx = fmaf(xv, xv, sxx);
          syy = fmaf(yv, yv, syy);
          sxy = fmaf(xv, yv, sxy);
          if (dr == 1 && dc == 1) { xc = xv; yc = yv; }
        }
      }
      const float inv9 = 1.f / 9.f;
      const float mx = sx * inv9, my = sy * inv9;
      const float sgx  = fmaxf(sxx * inv9 - mx * mx, 0.f);
      const float sgy  = fmaxf(syy * inv9 - my * my, 0.f);
      const float sgxy = sxy * inv9 - mx * my;
      const float C1v = 0.0001f, C2v = 0.0009f;
      const float nume = (2.f * mx * my + C1v) * (2.f * sgxy + C2v);
      const float deno = (mx * mx + my * my + C1v) * (sgx + sgy + C2v);
      const float ssim = fminf(fmaxf(0.5f * (1.f - nume / deno), 0.f), 1.f);
      const float l1   = fabsf(xc - yc);
      th_photo += vs[p] * ((0.85f / 3.f) * ssim + (0.15f / 3.f) * l1);
    }
  }

  // ---- Block reduction + one f64 atomic pair per tile.
  pr[tid] = th_photo;
  vr[tid] = th_valid;
  __syncthreads();
  for (int s = NT / 2; s > 0; s >>= 1) {
    if (tid < s) { pr[tid] += pr[tid + s]; vr[tid] += vr[tid + s]; }
    __syncthreads();
  }
  if (tid == 0) {
    atomicAdd(&ws[0], (double)pr[0]);
    atomicAdd(&ws[1], (double)vr[0]);
  }
}

__global__ void photo_final(const double* __restrict__ ws, float* __restrict__ out) {
  out[0] = (float)(ws[0] / fmax(ws[1], 1.0));
}

extern "C" void kernel_launch(void* const* d_in, const int* in_sizes, int n_in,
                              void* d_out, int out_size, void* d_ws, size_t ws_size,
                              hipStream_t stream) {
  (void)in_sizes; (void)n_in; (void)out_size; (void)ws_size;
  const float* disp  = (const float*)d_in[0];
  const float* left  = (const float*)d_in[1];
  const float* right = (const float*)d_in[2];
  float*  out = (float*)d_out;
  double* ws  = (double*)d_ws;

  photo_init<<<1, 64, 0, stream>>>(ws);
  photo_main<<<dim3(IMW / TW, IMH / TH, BATCH), NT, 0, stream>>>(disp, left, right, ws);
  photo_final<<<1, 1, 0, stream>>>(ws, out);
}